// GlobalFilter_33036888441334
// MI455X (gfx1250) — compile-verified
//
#include <hip/hip_runtime.h>

// GlobalFilter (rfft2 * complex_weight -> irfft2) on MI455X / gfx1250.
// One wave (32 lanes) processes one (batch, 16-channel) tile fully fused.
// All 14-point DFT stages run on V_WMMA_F32_16X16X4_F32 (f32 in / f32 acc).
// x staging uses GLOBAL_LOAD_ASYNC_TO_LDS_B128 (ASYNCcnt) when available.

typedef float v2f __attribute__((ext_vector_type(2)));
typedef float v8f __attribute__((ext_vector_type(8)));
typedef int   v4i_ __attribute__((vector_size(16)));              // int4 for b128
typedef __attribute__((address_space(1))) v4i_* gv4p;             // global int4*
typedef __attribute__((address_space(3))) v4i_* lv4p;             // lds int4*

#if __has_builtin(__builtin_amdgcn_global_load_async_to_lds_b128)
#define HAVE_ASYNC_LDS 1
#endif

// cos(pi*m/7), sin(pi*m/7) for m = 0..13  (2*pi*j*k/14 = pi*(j*k)/7, index (j*k)%14)
__constant__ float KCOS[14] = {
     1.0f,                 0.9009688679024191f,  0.6234898018587336f,
     0.22252093395631445f, -0.22252093395631434f, -0.6234898018587335f,
    -0.900968867902419f,  -1.0f,                 -0.9009688679024191f,
    -0.6234898018587336f, -0.22252093395631456f,  0.22252093395631423f,
     0.6234898018587334f,  0.900968867902419f };
__constant__ float KSIN[14] = {
     0.0f,                 0.4338837391175581f,  0.7818314824680298f,
     0.9749279121818236f,  0.9749279121818236f,  0.7818314824680299f,
     0.43388373911755823f, 0.0f,                -0.433883739117558f,
    -0.7818314824680297f, -0.9749279121818236f, -0.9749279121818236f,
    -0.78183148246803f,   -0.4338837391175583f };

#define NRM 0.26726124191242440f  /* 1/sqrt(14): ortho norm per 1D transform */

__device__ __forceinline__ v8f wmma4(v2f a, v2f b, v8f c) {
  return __builtin_amdgcn_wmma_f32_16x16x4_f32(false, a, false, b, (short)0, c,
                                               false, false);
}

// Build 4 A-fragments (K=16 split into 4 k-steps) for matrix f(m,k).
// f32 A 16x4 layout: lanes 0-15 -> M=lane, {K=4s, 4s+1}; lanes 16-31 -> M=lane-16, {K=4s+2, 4s+3}.
template <typename F>
__device__ __forceinline__ void build_frags(v2f* dst, int lane, F f) {
  const int m = lane & 15;
  const int ko = (lane & 16) ? 2 : 0;
#pragma unroll
  for (int s = 0; s < 4; ++s) {
    v2f a;
    a.x = f(m, 4 * s + ko);
    a.y = f(m, 4 * s + ko + 1);
    dst[s] = a;
  }
}

__global__ __launch_bounds__(32) void gf_wmma_kernel(
    const float* __restrict__ x,   // [256,196,512]
    const float* __restrict__ w,   // [14,8,512,2]
    float* __restrict__ out) {     // [256,196,512]
  // Region1 (4096 f): X[16(n2) x 14(n1) x 16(c)] (3584 incl. K-pad), reused as HRE(2048)+HIM(2048)
  // Region2 (4096 f): BRE[8(k2) x 14(n1) x 16(c)] pad->2048, BIM same; reused as TRE(1792)+TIM(1792)
  __shared__ float sm[8192];
  float* const R1 = sm;
  float* const R2 = sm + 4096;

  const int lane = (int)threadIdx.x;           // 0..31
  const int b    = (int)blockIdx.x >> 5;       // batch
  const int c0   = ((int)blockIdx.x & 31) << 4;// channel tile base

  const float* xb = x + (size_t)b * 196 * 512 + c0;

#ifdef HAVE_ASYNC_LDS
  // ---- async stage x[b, :, c0:c0+16] straight into LDS as X[n2][n1][c] ----
  // Per-lane 16B global->LDS copies, no VGPR round trip (ASYNCcnt tracked).
  // AS3 pointer = low 32 bits of the flat address (wave-relative LDS offset).
  for (int j = lane; j < 784; j += 32) {           // 196 rows * 4 float4
    const int row = j >> 2, cp = (j & 3) << 2;
    const int n1 = row / 14, n2 = row % 14;        // n = n1*14 + n2
    __builtin_amdgcn_global_load_async_to_lds_b128(
        (gv4p)(uintptr_t)(xb + (size_t)row * 512 + cp),
        (lv4p)(uintptr_t)(R1 + ((n2 * 14 + n1) * 16 + cp)),
        0, 0);
  }
#endif

  // ---- zero K-padding regions (guards 0*garbage against NaN) ----
  for (int i = lane; i < 448; i += 32) R1[3136 + i] = 0.0f;        // X rows n2=14,15
  for (int i = lane; i < 256; i += 32) { R2[1792 + i] = 0.0f;      // BRE tail
                                         R2[3840 + i] = 0.0f; }    // BIM tail

  // ---- warm L2 with this unit's weight slice (reused by all 256 batches) ----
  for (int i = lane; i < 112; i += 32)             // 14*8 frequency rows
    __builtin_prefetch(w + ((size_t)i * 512 + c0) * 2, 0, 0);

#ifndef HAVE_ASYNC_LDS
  // ---- fallback staging through VGPRs (float4 loads + ds stores) ----
  for (int j = lane; j < 784; j += 32) {
    const int row = j >> 2, cp = (j & 3) << 2;
    const int n1 = row / 14, n2 = row % 14;
    const float4 v = *(const float4*)(xb + (size_t)row * 512 + cp);
    *(float4*)(R1 + ((n2 * 14 + n1) * 16 + cp)) = v;
  }
#endif

  // ---- build all transform-matrix A fragments (zero outside valid range) ----
  v2f aM1r[4], aM1i[4], aC2[4], aS2[4], aS2n[4], aA4r[4], aA4i[4];
  build_frags(aM1r, lane, [](int m, int k) {        // rfft rows k2<8, cols n2<14
    return (m < 8 && k < 14) ? KCOS[(m * k) % 14] * NRM : 0.0f; });
  build_frags(aM1i, lane, [](int m, int k) {
    return (m < 8 && k < 14) ? -KSIN[(m * k) % 14] * NRM : 0.0f; });
  build_frags(aC2, lane, [](int m, int k) {         // full 14x14 cos / sin
    return (m < 14 && k < 14) ? KCOS[(m * k) % 14] * NRM : 0.0f; });
  build_frags(aS2, lane, [](int m, int k) {
    return (m < 14 && k < 14) ? KSIN[(m * k) % 14] * NRM : 0.0f; });
  build_frags(aS2n, lane, [](int m, int k) {
    return (m < 14 && k < 14) ? -KSIN[(m * k) % 14] * NRM : 0.0f; });
  build_frags(aA4r, lane, [](int m, int k) {        // c2r fold: alpha * cos/-sin
    if (m >= 14 || k >= 8) return 0.0f;
    const float a = (k == 0 || k == 7) ? 1.0f : 2.0f;
    return a * KCOS[(m * k) % 14] * NRM; });
  build_frags(aA4i, lane, [](int m, int k) {
    if (m >= 14 || k >= 8) return 0.0f;
    const float a = (k == 0 || k == 7) ? 1.0f : 2.0f;
    return -a * KSIN[(m * k) % 14] * NRM; });

  const int n  = lane & 15;
  const int ko = (lane & 16) ? 2 : 0;
  const int r0 = (lane & 16) ? 8 : 0;

#ifdef HAVE_ASYNC_LDS
  // All async copies must land in LDS before T1 reads fragments.
#if __has_builtin(__builtin_amdgcn_s_wait_asynccnt)
  __builtin_amdgcn_s_wait_asynccnt(0);
#else
  asm volatile("s_wait_asynccnt 0x0" ::: "memory");
#endif
  asm volatile("" ::: "memory");   // keep LDS reads below the wait
#endif

  // ================= T1: rfft along n2.  R[k2][n1][c] = M1 * X =================
  for (int t = 0; t < 14; ++t) {                 // t = n1
    v8f ar = {0,0,0,0,0,0,0,0}, ai = {0,0,0,0,0,0,0,0};
#pragma unroll
    for (int s = 0; s < 4; ++s) {
      const int kb = 4 * s + ko;
      v2f bx; bx.x = R1[(kb * 14 + t) * 16 + n];
              bx.y = R1[((kb + 1) * 14 + t) * 16 + n];
      ar = wmma4(aM1r[s], bx, ar);
      ai = wmma4(aM1i[s], bx, ai);
    }
    if (lane < 16) {
#pragma unroll
      for (int r = 0; r < 8; ++r) {              // rows k2 = 0..7 only
        R2[(r * 14 + t) * 16 + lane]        = ar[r];   // BRE
        R2[2048 + (r * 14 + t) * 16 + lane] = ai[r];   // BIM
      }
    }
  }

  // X is dead: region1 becomes HRE/HIM. Zero their K-pad rows (k1 = 14,15).
  for (int i = lane; i < 256; i += 32) { R1[1792 + i] = 0.0f; R1[3840 + i] = 0.0f; }

  // ====== T2: fft along n1, fused complex-weight multiply. H[k1][k2][c] ======
  for (int t = 0; t < 8; ++t) {                  // t = k2
    v8f fr = {0,0,0,0,0,0,0,0}, fi = {0,0,0,0,0,0,0,0};
#pragma unroll
    for (int s = 0; s < 4; ++s) {
      const int kb = 4 * s + ko;                 // k = n1
      v2f bre; bre.x = R2[(t * 14 + kb) * 16 + n];
               bre.y = R2[(t * 14 + kb + 1) * 16 + n];
      v2f bim; bim.x = R2[2048 + (t * 14 + kb) * 16 + n];
               bim.y = R2[2048 + (t * 14 + kb + 1) * 16 + n];
      fr = wmma4(aC2[s],  bre, fr);   // Ff_re = C*R_re + S*R_im
      fr = wmma4(aS2[s],  bim, fr);
      fi = wmma4(aC2[s],  bim, fi);   // Ff_im = C*R_im - S*R_re
      fi = wmma4(aS2n[s], bre, fi);
    }
#pragma unroll
    for (int r = 0; r < 8; ++r) {
      const int m = r0 + r;                      // m = k1
      if (m < 14) {
        const float2 wv = *(const float2*)(w + ((size_t)(m * 8 + t) * 512 + c0 + n) * 2);
        const float hre = fr[r] * wv.x - fi[r] * wv.y;
        const float him = fr[r] * wv.y + fi[r] * wv.x;
        R1[(m * 8 + t) * 16 + n]        = hre;   // HRE
        R1[2048 + (m * 8 + t) * 16 + n] = him;   // HIM
      }
    }
  }

  // ================= S3: ifft along k1.  T[k2][n1][c] =================
  for (int t = 0; t < 8; ++t) {                  // t = k2
    v8f tr = {0,0,0,0,0,0,0,0}, ti = {0,0,0,0,0,0,0,0};
#pragma unroll
    for (int s = 0; s < 4; ++s) {
      const int kb = 4 * s + ko;                 // k = k1
      v2f bhr; bhr.x = R1[(kb * 8 + t) * 16 + n];
               bhr.y = R1[((kb + 1) * 8 + t) * 16 + n];
      v2f bhi; bhi.x = R1[2048 + (kb * 8 + t) * 16 + n];
               bhi.y = R1[2048 + ((kb + 1) * 8 + t) * 16 + n];
      tr = wmma4(aC2[s],  bhr, tr);   // T_re = C*H_re - S*H_im
      tr = wmma4(aS2n[s], bhi, tr);
      ti = wmma4(aC2[s],  bhi, ti);   // T_im = C*H_im + S*H_re
      ti = wmma4(aS2[s],  bhr, ti);
    }
#pragma unroll
    for (int r = 0; r < 8; ++r) {
      const int m = r0 + r;                      // m = n1
      if (m < 14) {
        R2[(t * 14 + m) * 16 + n]        = tr[r];   // TRE
        R2[1792 + (t * 14 + m) * 16 + n] = ti[r];   // TIM
      }
    }
  }

  // ============ S4: c2r along k2, direct global store of y ============
  float* ob = out + (size_t)b * 196 * 512 + c0;
  for (int t = 0; t < 14; ++t) {                 // t = n1
    v8f y = {0,0,0,0,0,0,0,0};
#pragma unroll
    for (int s = 0; s < 2; ++s) {                // K = 8 (k2)
      const int kb = 4 * s + ko;
      v2f btr; btr.x = R2[(kb * 14 + t) * 16 + n];
               btr.y = R2[((kb + 1) * 14 + t) * 16 + n];
      v2f bti; bti.x = R2[1792 + (kb * 14 + t) * 16 + n];
               bti.y = R2[1792 + ((kb + 1) * 14 + t) * 16 + n];
      y = wmma4(aA4r[s], btr, y);     // y = A4r*T_re + A4i*T_im
      y = wmma4(aA4i[s], bti, y);
    }
#pragma unroll
    for (int r = 0; r < 8; ++r) {
      const int m = r0 + r;                      // m = n2
      if (m < 14) ob[(size_t)(t * 14 + m) * 512 + n] = y[r];
    }
  }
}

extern "C" void kernel_launch(void* const* d_in, const int* in_sizes, int n_in,
                              void* d_out, int out_size, void* d_ws, size_t ws_size,
                              hipStream_t stream) {
  const float* x = (const float*)d_in[0];        // [B,196,512] f32
  const float* w = (const float*)d_in[1];        // [14,8,512,2] f32
  float* out = (float*)d_out;                    // [B,196,512] f32
  const int B = in_sizes[0] / (196 * 512);       // 256
  const int nblocks = B * 32;                    // 32 channel tiles of 16
  gf_wmma_kernel<<<dim3(nblocks), dim3(32), 0, stream>>>(x, w, out);
}